// Agent_81758997446938
// MI455X (gfx1250) — compile-verified
//
#include <hip/hip_runtime.h>
#include <hip/hip_bf16.h>

typedef float v2f __attribute__((ext_vector_type(2)));
typedef float v8f __attribute__((ext_vector_type(8)));
typedef int   v4i __attribute__((ext_vector_type(4)));
typedef __attribute__((address_space(1))) v4i* gptr4;   // global int4*
typedef __attribute__((address_space(3))) v4i* lptr4;   // LDS int4*

#define LOG2PI 1.8378770664093453f
#define EPSF   1.1920928955078125e-07f

#define XS 132           // x tile stride: bank = (4r+c)%64 -> conflict-free, 16B-aligned rows
#define HS 68            // hidden scratch stride
#define RS 37            // raw (TRIL=36) scratch stride
#define WAVE_F (16*XS + 16*HS + 16*RS + 16*8 + 16*4)   // 3984 floats / wave (63744 B total)

static __device__ __forceinline__ v8f wmma4(v2f a, v2f b, v8f c) {
  // V_WMMA_F32_16X16X4_F32 : D = A(16x4,f32) x B(4x16,f32) + C(16x16,f32)
  return __builtin_amdgcn_wmma_f32_16x16x4_f32(false, a, false, b, (short)0, c,
                                               false, false);
}

static __device__ __forceinline__ float silu_f(float v) {
  // v * sigmoid(v); v_rcp_f32 instead of the IEEE div sequence (1-ulp)
  return v * __builtin_amdgcn_rcpf(1.0f + __expf(-v));
}

// D(16xN) = act(A(16xK, LDS stride As) @ W(KxN, global row-major) + bias)
// NT = ceil(N/16) <= 4. Accumulators fully buffered before the store phase, so
// D may alias A (DS ops are in-order within a wave).
static __device__ __forceinline__ void gemm_lds(
    const float* A, int As, int K,
    const float* __restrict__ W, int N, int NT,
    const float* __restrict__ bias,
    float* D, int Ds, bool do_silu, int lane) {
  const int row = lane & 15;
  const int ko  = (lane >> 4) << 1;      // K sub-offset 0 or 2 (ISA A/B layout)
  const v8f zero = {0,0,0,0,0,0,0,0};
  v8f acc[4];
  #pragma unroll
  for (int t = 0; t < NT; ++t) acc[t] = zero;
  for (int k0 = 0; k0 < K; k0 += 4) {
    v2f a;
    a.x = A[row*As + k0 + ko];
    a.y = A[row*As + k0 + ko + 1];
    #pragma unroll
    for (int t = 0; t < NT; ++t) {
      int n = t*16 + row;
      bool ok = (n < N);
      v2f b;
      b.x = ok ? W[(k0+ko  )*N + n] : 0.f;
      b.y = ok ? W[(k0+ko+1)*N + n] : 0.f;
      acc[t] = wmma4(a, b, acc[t]);
    }
  }
  const int mb8 = (lane >> 4) << 3;      // C/D layout: lanes 16-31 hold M=8..15
  #pragma unroll
  for (int t = 0; t < NT; ++t) {
    int n = t*16 + row;
    if (n < N) {
      float bv = bias[n];
      #pragma unroll
      for (int r = 0; r < 8; ++r) {
        float v = acc[t][r] + bv;
        if (do_silu) v = silu_f(v);
        D[(mb8 + r)*Ds + n] = v;
      }
    }
  }
}

// flow layer 1: st1 = silu([cond(4) | x(128)] @ W(132x32) + b), into D (stride HS)
static __device__ __forceinline__ void flow_gemm1(
    const float* cnd, const float* xs,
    const float* __restrict__ W, const float* __restrict__ bias,
    float* D, int lane) {
  const int row = lane & 15;
  const int ko  = (lane >> 4) << 1;
  const v8f zero = {0,0,0,0,0,0,0,0};
  v8f acc0 = zero, acc1 = zero;
  {  // K block 0..3 : conditioning values
    v2f a; a.x = cnd[row*4 + ko]; a.y = cnd[row*4 + ko + 1];
    v2f b0, b1;
    b0.x = W[ko*32 + row];       b0.y = W[(ko+1)*32 + row];
    b1.x = W[ko*32 + 16 + row];  b1.y = W[(ko+1)*32 + 16 + row];
    acc0 = wmma4(a, b0, acc0);
    acc1 = wmma4(a, b1, acc1);
  }
  for (int k0 = 4; k0 < 132; k0 += 4) {   // K blocks 4..131 : x columns
    v2f a;
    a.x = xs[row*XS + (k0-4) + ko];
    a.y = xs[row*XS + (k0-4) + ko + 1];
    v2f b0, b1;
    b0.x = W[(k0+ko)*32 + row];       b0.y = W[(k0+ko+1)*32 + row];
    b1.x = W[(k0+ko)*32 + 16 + row];  b1.y = W[(k0+ko+1)*32 + 16 + row];
    acc0 = wmma4(a, b0, acc0);
    acc1 = wmma4(a, b1, acc1);
  }
  const int mb8 = (lane >> 4) << 3;
  float bv0 = bias[row], bv1 = bias[16 + row];
  #pragma unroll
  for (int r = 0; r < 8; ++r) {
    D[(mb8 + r)*HS + row]      = silu_f(acc0[r] + bv0);
    D[(mb8 + r)*HS + 16 + row] = silu_f(acc1[r] + bv1);
  }
}

// row-wise RMS norm over N columns; both wave halves cooperate (row = lane&15,
// half = lane>>4 handles columns [half*N/2, half*N/2+N/2)), combined via shfl_xor.
static __device__ __forceinline__ void rms_rows(float* H, int Hs, int N,
                                                const float* __restrict__ w,
                                                int lane) {
  const int row  = lane & 15;
  const int c0   = (lane >> 4) * (N >> 1);
  float ss = 0.f;
  #pragma unroll 8
  for (int n = 0; n < (N >> 1); ++n) {
    float v = H[row*Hs + c0 + n];
    ss += v*v;
  }
  ss += __shfl_xor(ss, 16, 32);
  float sc = rsqrtf(ss / (float)N + EPSF);
  #pragma unroll 8
  for (int n = 0; n < (N >> 1); ++n) {
    int c = c0 + n;
    H[row*Hs + c] = H[row*Hs + c] * sc * w[c];
  }
}

__global__ __launch_bounds__(128) void Agent_81758997446938_kernel(
    const float* __restrict__ x, const float* __restrict__ action,
    const float* __restrict__ z_rpo,
    const float* __restrict__ cW1, const float* __restrict__ cb1,
    const float* __restrict__ c_rms,
    const float* __restrict__ cW2, const float* __restrict__ cb2,
    const float* __restrict__ cW3, const float* __restrict__ cb3,
    const float* __restrict__ aW1, const float* __restrict__ ab1,
    const float* __restrict__ aW2, const float* __restrict__ ab2,
    const float* __restrict__ mW,  const float* __restrict__ mb,
    const float* __restrict__ tW1, const float* __restrict__ tb1,
    const float* __restrict__ t_rms1,
    const float* __restrict__ tW2, const float* __restrict__ tb2,
    const float* __restrict__ t_rms2,
    const float* __restrict__ hW,  const float* __restrict__ hb,
    const float* __restrict__ fW1, const float* __restrict__ fb1,
    const float* __restrict__ fW2, const float* __restrict__ fb2,
    float* __restrict__ out, int Bn)
{
  __shared__ float smem[4 * WAVE_F];            // ~62.3 KB static LDS
  const int lane = threadIdx.x & 31;
  const int wave = threadIdx.x >> 5;
  const int row0 = blockIdx.x * 64 + wave * 16;
  if (row0 >= Bn) return;

  float* base = smem + wave * WAVE_F;
  float* xs  = base;                  // 16 x 128 (stride XS)
  float* hsc = xs  + 16*XS;           // 16 x 64/32 hidden scratch (stride HS)
  float* rsc = hsc + 16*HS;           // 16 x 36 raw tril (stride RS)
  float* rb  = rsc + 16*RS;           // 16 x 8 (action_mean / flow st)
  float* cnd = rb  + 16*8;            // 16 x 4 conditioning values

  float* out_act = out;
  float* out_lp  = out + (size_t)Bn * 8;
  float* out_en  = out + (size_t)Bn * 9;
  float* out_v   = out + (size_t)Bn * 10;

  if (threadIdx.x == 0) {             // warm L2 paths (global_prefetch_b8)
    __builtin_prefetch(aW1, 0, 0);
    __builtin_prefetch(tW1, 0, 0);
    __builtin_prefetch(cW1, 0, 0);
  }

  // ---- stage x tile into LDS + pass action through
  {
    const float* xg = x + (size_t)row0 * 128;
#if defined(__gfx1250__) && __has_builtin(__builtin_amdgcn_global_load_async_to_lds_b128)
    // async global->LDS DMA path: no VGPR staging, tracked by ASYNCcnt
    #pragma unroll
    for (int j = 0; j < 16; ++j) {
      int flat = lane*4 + j*128;
      float* d = xs + (flat >> 7) * XS + (flat & 127);    // 16B-aligned (XS=132)
      __builtin_amdgcn_global_load_async_to_lds_b128(
          (gptr4)(xg + flat), (lptr4)d, 0, 0);
    }
    asm volatile("s_wait_asynccnt 0x0" ::: "memory");
#else
    #pragma unroll
    for (int j = 0; j < 16; ++j) {
      int flat = lane*4 + j*128;
      const float4 v = *(const float4*)(xg + flat);
      *(float4*)(xs + (flat >> 7) * XS + (flat & 127)) = v;
    }
#endif
    const float* ag = action + (size_t)row0 * 8;
    float* og = out_act + (size_t)row0 * 8;
    #pragma unroll
    for (int j = 0; j < 4; ++j) og[lane + 32*j] = ag[lane + 32*j];
  }

  // ---- t-branch: raw = rms(silu(rms(silu(x@tW1+b),w1)@tW2+b),w2) @ hW + hb
  gemm_lds(xs,  XS, 128, tW1, 64, 4, tb1, hsc, HS, true,  lane);
  rms_rows(hsc, HS, 64, t_rms1, lane);
  gemm_lds(hsc, HS,  64, tW2, 64, 4, tb2, hsc, HS, true,  lane);
  rms_rows(hsc, HS, 64, t_rms2, lane);
  gemm_lds(hsc, HS,  64, hW,  36, 3, hb,  rsc, RS, false, lane);

  // ---- actor mean: am = silu(silu(x@aW1+b)@aW2+b)@mW + mb
  gemm_lds(xs,  XS, 128, aW1, 64, 4, ab1, hsc, HS, true,  lane);
  gemm_lds(hsc, HS,  64, aW2, 64, 4, ab2, hsc, HS, true,  lane);
  gemm_lds(hsc, HS,  64, mW,   8, 1, mb,  rb,  8,  false, lane);

  // ---- per-row noise init: xr = action - (am + z_rpo)
  float xr[8];
  float fld = 0.f;
  if (lane < 16) {
    const float* ag = action + (size_t)(row0 + lane) * 8;
    const float* zg = z_rpo  + (size_t)(row0 + lane) * 8;
    #pragma unroll
    for (int j = 0; j < 8; ++j) xr[j] = ag[j] - (rb[lane*8 + j] + zg[j]);
  }

  // ---- inverse flow, layers 3..0
  for (int i = 3; i >= 0; --i) {
    const bool even = (i & 1) == 0;
    if (lane < 16) {
      #pragma unroll
      for (int j = 0; j < 4; ++j) cnd[lane*4 + j] = even ? xr[j] : xr[4 + j];
    }
    flow_gemm1(cnd, xs, fW1 + i*132*32, fb1 + i*32, hsc, lane);
    gemm_lds(hsc, HS, 32, fW2 + i*32*8, 8, 1, fb2 + i*8, rb, 8, false, lane);
    if (lane < 16) {
      #pragma unroll
      for (int j = 0; j < 4; ++j) {
        float sraw = rb[lane*8 + j];
        float tsh  = rb[lane*8 + 4 + j];
        float s    = 2.0f * tanhf(sraw);
        float y    = even ? xr[4 + j] : xr[j];
        float xt   = (y - tsh) * __expf(-s);
        if (even) xr[4 + j] = xt; else xr[j] = xt;
        fld += s;
      }
    }
  }

  // ---- softplus diag, log-det, forward substitution, log-prob / entropy
  if (lane < 16) {
    const int rg = row0 + lane;
    float diag[8];
    float logdet = 0.f;
    #pragma unroll
    for (int i = 0; i < 8; ++i) {
      float r  = rsc[lane*RS + (i*(i+1))/2 + i];
      float sp = fmaxf(r, 0.f) + log1pf(__expf(-fabsf(r)));   // stable softplus
      diag[i]  = sp;
      logdet  += __logf(sp);
    }
    float z[8];
    float mahal = 0.f;
    #pragma unroll
    for (int i = 0; i < 8; ++i) {
      float s = xr[i];                      // correlated = [x1|x2] = xr
      #pragma unroll
      for (int j = 0; j < i; ++j) s -= rsc[lane*RS + (i*(i+1))/2 + j] * z[j];
      z[i] = s / diag[i];
      mahal += z[i]*z[i];
    }
    out_lp[rg] = -0.5f * (8.0f*LOG2PI + mahal) - logdet - fld;
    out_en[rg] =  0.5f * (8.0f*(1.0f + LOG2PI)) + logdet + fld;
  }

  // ---- critic: v = silu(rms(silu(x@cW1+b),w)@cW2+b) @ cW3 + cb3
  gemm_lds(xs,  XS, 128, cW1, 64, 4, cb1, hsc, HS, true, lane);
  rms_rows(hsc, HS, 64, c_rms, lane);
  gemm_lds(hsc, HS,  64, cW2, 64, 4, cb2, hsc, HS, true, lane);
  {
    const int row = lane & 15;
    const int c0  = (lane >> 4) * 32;
    float s = 0.f;
    #pragma unroll 8
    for (int k = 0; k < 32; ++k) s += hsc[row*HS + c0 + k] * cW3[c0 + k];
    s += __shfl_xor(s, 16, 32);
    if (lane < 16) out_v[row0 + row] = s + cb3[0];
  }
}

extern "C" void kernel_launch(void* const* d_in, const int* in_sizes, int n_in,
                              void* d_out, int out_size, void* d_ws, size_t ws_size,
                              hipStream_t stream) {
  const float* x      = (const float*)d_in[0];
  const float* action = (const float*)d_in[1];
  const float* z_rpo  = (const float*)d_in[2];
  const float* cW1 = (const float*)d_in[3];
  const float* cb1 = (const float*)d_in[4];
  const float* c_rms = (const float*)d_in[5];
  const float* cW2 = (const float*)d_in[6];
  const float* cb2 = (const float*)d_in[7];
  const float* cW3 = (const float*)d_in[8];
  const float* cb3 = (const float*)d_in[9];
  const float* aW1 = (const float*)d_in[10];
  const float* ab1 = (const float*)d_in[11];
  const float* aW2 = (const float*)d_in[12];
  const float* ab2 = (const float*)d_in[13];
  const float* mW  = (const float*)d_in[14];
  const float* mb  = (const float*)d_in[15];
  const float* tW1 = (const float*)d_in[16];
  const float* tb1 = (const float*)d_in[17];
  const float* t_rms1 = (const float*)d_in[18];
  const float* tW2 = (const float*)d_in[19];
  const float* tb2 = (const float*)d_in[20];
  const float* t_rms2 = (const float*)d_in[21];
  const float* hW  = (const float*)d_in[22];
  const float* hb  = (const float*)d_in[23];
  const float* fW1 = (const float*)d_in[24];
  const float* fb1 = (const float*)d_in[25];
  const float* fW2 = (const float*)d_in[26];
  const float* fb2 = (const float*)d_in[27];

  const int Bn = in_sizes[0] / 128;
  const int grid = (Bn + 63) / 64;            // 64 rows per block (4 waves x 16)

  Agent_81758997446938_kernel<<<dim3(grid), dim3(128), 0, stream>>>(
      x, action, z_rpo,
      cW1, cb1, c_rms, cW2, cb2, cW3, cb3,
      aW1, ab1, aW2, ab2, mW, mb,
      tW1, tb1, t_rms1, tW2, tb2, t_rms2,
      hW, hb, fW1, fb1, fW2, fb2,
      (float*)d_out, Bn);
}